// GATLayer_52785148068150
// MI455X (gfx1250) — compile-verified
//
#include <hip/hip_runtime.h>
#include <hip/hip_bf16.h>

typedef __attribute__((ext_vector_type(2))) float v2f;
typedef __attribute__((ext_vector_type(8))) float v8f;

#define D_IN  128
#define D_OUT 128
#define WPB   8   // waves per 256-thread block (wave32)

// ---------------------------------------------------------------------------
// Kernel 1 (fused): one wave computes a full 16x128 row band of h = x @ W^T
// with 8 independent V_WMMA_F32_16X16X4_F32 accumulator chains (N-blocking),
// then computes s = h . a_src and d = h . a_dst in-register.
//
// A fragment (ISA 7.12.2, 32-bit A 16x4): lanes 0-15 hold K={0,1} in the two
// A VGPRs, lanes 16-31 hold K={2,3} -> contiguous per-lane float2 from x.
// A is loaded ONCE per k-step and reused by all 8 col tiles (x traffic = 25.6MB
// minimum instead of 8x redundant). B[k][n] = W[n][k] -> contiguous float2
// from row n of W; W (64KB) serves from L1/L2.
// C/D layout: VGPR r, lanes 0-15 -> M=r, lanes 16-31 -> M=r+8; so row m<8
// lives entirely in lanes 0-15 and row m>=8 in lanes 16-31: the s/d reduction
// is a 16-lane __shfl_xor tree that never crosses the half-wave boundary.
// ---------------------------------------------------------------------------
__global__ __launch_bounds__(256) void gat_gemm_fused(
    const float* __restrict__ x, const float* __restrict__ W,
    const float* __restrict__ a, float* __restrict__ h,
    float* __restrict__ s, float* __restrict__ d, int nNodes) {
  const int wave = blockIdx.x * WPB + (threadIdx.x >> 5);
  const int lane = threadIdx.x & 31;
  const int rowBase = wave * 16;
  if (rowBase >= nNodes) return;                  // wave-uniform; EXEC stays full

  const int lr   = lane & 15;
  const int half = lane >> 4;                     // 0 -> K+{0,1}, 1 -> K+{2,3}

  const float* arow = x + (size_t)(rowBase + lr) * D_IN + 2 * half;
  const float* brow = W + (size_t)lr * D_IN + 2 * half;   // col tile c: + c*16*D_IN

  v8f acc[8] = {};                                // 8 independent WMMA chains
  #pragma unroll
  for (int k = 0; k < D_IN; k += 4) {
    const v2f a2 = *(const v2f*)(arow + k);       // one A load, reused 8x
    #pragma unroll
    for (int c = 0; c < 8; ++c) {
      const v2f b2 = *(const v2f*)(brow + (size_t)c * 16 * D_IN + k);
      acc[c] = __builtin_amdgcn_wmma_f32_16x16x4_f32(
          /*neg_a=*/false, a2, /*neg_b=*/false, b2,
          /*c_mod=*/(short)0, acc[c], /*reuse_a=*/false, /*reuse_b=*/false);
    }
  }

  // Per-lane slice of the attention vectors: column c*16+lr for c = 0..7
  float av[8], bv[8];
  #pragma unroll
  for (int c = 0; c < 8; ++c) {
    av[c] = a[c * 16 + lr];
    bv[c] = a[D_OUT + c * 16 + lr];
  }

  #pragma unroll
  for (int r = 0; r < 8; ++r) {
    const int m = rowBase + r + 8 * half;         // global node row for this VGPR
    float sp = 0.f, dp = 0.f;
    #pragma unroll
    for (int c = 0; c < 8; ++c) {
      h[(size_t)m * D_OUT + c * 16 + lr] = acc[c][r];
      sp += acc[c][r] * av[c];
      dp += acc[c][r] * bv[c];
    }
    #pragma unroll
    for (int off = 8; off > 0; off >>= 1) {       // reduce within the 16-lane half
      sp += __shfl_xor(sp, off, 32);
      dp += __shfl_xor(dp, off, 32);
    }
    if (lr == 0) { s[m] = sp; d[m] = dp; }
  }
}

// ---------------------------------------------------------------------------
// Kernel 2: zero the output accumulator (harness poisons d_out with 0xAA).
// ---------------------------------------------------------------------------
__global__ __launch_bounds__(256) void gat_zero(float4* __restrict__ f, int nVec4) {
  const int i = blockIdx.x * blockDim.x + threadIdx.x;
  if (i < nVec4) f[i] = make_float4(0.f, 0.f, 0.f, 0.f);
}

// ---------------------------------------------------------------------------
// Kernel 3: per-edge  e = sigmoid(s[src]+d[dst]);  f[dst,:] += e * h[src,:]
// One wave32 per edge; float4 gather; 4 hardware f32 atomic adds per lane.
// ~820 MB of gather+scatter traffic dominates the whole op; the 25.6 MB f
// array is L2-resident (192 MB), so the atomics resolve on-chip.
// ---------------------------------------------------------------------------
__global__ __launch_bounds__(256) void gat_edges(
    const float* __restrict__ h, const float* __restrict__ s,
    const float* __restrict__ d, const int* __restrict__ src,
    const int* __restrict__ dst, float* __restrict__ f, int nEdges) {
  const int edge = blockIdx.x * WPB + (threadIdx.x >> 5);
  if (edge >= nEdges) return;
  const int lane = threadIdx.x & 31;

  const int si = src[edge];
  const int di = dst[edge];
  const float z = s[si] + d[di];
  const float e = 1.0f / (1.0f + __expf(-z));     // sigmoid

  const float4 hv = ((const float4*)(h + (size_t)si * D_OUT))[lane];
  float* frow = f + (size_t)di * D_OUT + lane * 4;
  unsafeAtomicAdd(frow + 0, e * hv.x);            // global_atomic_add_f32
  unsafeAtomicAdd(frow + 1, e * hv.y);
  unsafeAtomicAdd(frow + 2, e * hv.z);
  unsafeAtomicAdd(frow + 3, e * hv.w);
}

// ---------------------------------------------------------------------------
// Launcher. Inputs: x(f32 N*128), W(f32 128*128), a(f32 256), src(i32 E),
// dst(i32 E). Output: f32 N*128. Workspace: h (N*128 f32), s (N), d (N).
// ---------------------------------------------------------------------------
extern "C" void kernel_launch(void* const* d_in, const int* in_sizes, int n_in,
                              void* d_out, int out_size, void* d_ws, size_t ws_size,
                              hipStream_t stream) {
  const float* x   = (const float*)d_in[0];
  const float* W   = (const float*)d_in[1];
  const float* a   = (const float*)d_in[2];
  const int*   src = (const int*)d_in[3];
  const int*   dst = (const int*)d_in[4];
  float* f = (float*)d_out;

  const int nNodes = in_sizes[0] / D_IN;   // 50000
  const int nEdges = in_sizes[3];          // 800000

  float* h = (float*)d_ws;                                   // nNodes*128 f32
  float* s = h + (size_t)nNodes * D_OUT;                     // nNodes f32
  float* d = s + nNodes;                                     // nNodes f32

  // 1) fused GEMM + attention scores: one wave per 16-row band
  {
    const int rowTiles = (nNodes + 15) / 16;                 // 3125
    const int blocks = (rowTiles + WPB - 1) / WPB;
    gat_gemm_fused<<<blocks, 256, 0, stream>>>(x, W, a, h, s, d, nNodes);
  }
  // 2) zero output
  {
    const int nVec4 = nNodes * (D_OUT / 4);
    const int blocks = (nVec4 + 255) / 256;
    gat_zero<<<blocks, 256, 0, stream>>>((float4*)f, nVec4);
  }
  // 3) edge scatter: one wave per edge
  {
    const int blocks = (nEdges + WPB - 1) / WPB;
    gat_edges<<<blocks, 256, 0, stream>>>(h, s, d, src, dst, f, nEdges);
  }
}